// GCNBlock_4887672783235
// MI455X (gfx1250) — compile-verified
//
#include <hip/hip_runtime.h>

#define N_NODES 100000
#define N_EDGES 640000
#define D       128
#define BN_EPS  1e-5f
#define LDS_PITCH 132   // 128 + 4 pad: breaks the 512B row stride bank conflict

typedef float v2f __attribute__((ext_vector_type(2)));
typedef float v8f __attribute__((ext_vector_type(8)));

// ---------------- degree / normalization ----------------
__global__ void k_init_deg(float* __restrict__ deg) {
    int i = blockIdx.x * blockDim.x + threadIdx.x;
    if (i < N_NODES) deg[i] = 1.0f;   // self-loop contributes 1
}

__global__ void k_deg(const int* __restrict__ dst, float* __restrict__ deg) {
    int e = blockIdx.x * blockDim.x + threadIdx.x;
    if (e < N_EDGES) atomicAdd(&deg[dst[e]], 1.0f);
}

__global__ void k_dinv(float* __restrict__ deg) {
    int i = blockIdx.x * blockDim.x + threadIdx.x;
    if (i < N_NODES) deg[i] = rsqrtf(deg[i]);   // deg >= 1, in-place -> dinv
}

// ---------------- h = x @ W via V_WMMA_F32_16X16X4_F32 ----------------
// Block: 256 threads = 8 waves. Block owns 16 rows of x; wave w owns columns
// [16w, 16w+16). K loop: 128/4 = 32 WMMA issues per wave.
__global__ void __launch_bounds__(256) k_gemm(const float* __restrict__ x,
                                              const float* __restrict__ Wm,
                                              float* __restrict__ h) {
    __shared__ float lx[16 * LDS_PITCH];
    const int tid  = threadIdx.x;
    const int row0 = blockIdx.x * 16;

    // cooperative stage of the 16x128 x-tile into LDS
    for (int idx = tid; idx < 16 * D; idx += 256) {
        int r = idx >> 7, c = idx & (D - 1);
        lx[r * LDS_PITCH + c] = x[(row0 + r) * D + c];
    }
    __syncthreads();

    const int wave = tid >> 5;
    const int lane = tid & 31;
    const int half = lane >> 4;      // 0: K={0,1}, 1: K={2,3}
    const int l15  = lane & 15;      // M index for A, N index for B
    const int koff = half * 2;
    const int n0   = wave * 16;      // this wave's output-column base

    v8f acc = {};
    for (int k = 0; k < D; k += 4) {
        v2f a, b;
        // A 16x4 (ISA 7.12.2): lanes 0-15 -> {A[m][k],A[m][k+1]}, lanes 16-31 -> K+2
        a.x = lx[l15 * LDS_PITCH + k + koff];
        a.y = lx[l15 * LDS_PITCH + k + koff + 1];
        // B 4x16: lanes encode N, VGPR index + lane-half encode K
        b.x = Wm[(k + koff)     * D + n0 + l15];
        b.y = Wm[(k + koff + 1) * D + n0 + l15];
        acc = __builtin_amdgcn_wmma_f32_16x16x4_f32(
            /*neg_a=*/false, a, /*neg_b=*/false, b,
            /*c_mod=*/(short)0, acc, /*reuse_a=*/false, /*reuse_b=*/false);
    }

    // D 16x16 f32 layout: VGPR r -> M=r (lanes 0-15) / M=r+8 (lanes 16-31)
#pragma unroll
    for (int r = 0; r < 8; ++r) {
        int row = row0 + r + half * 8;
        h[row * D + n0 + l15] = acc[r];
    }
}

// ---------------- out = b + h * dinv^2  (bias + self-loop message) ----------------
__global__ void k_self_bias(const float* __restrict__ h, const float* __restrict__ dinv,
                            const float* __restrict__ b, float* __restrict__ out) {
    int i = blockIdx.x * blockDim.x + threadIdx.x;
    if (i < N_NODES * D) {
        int n = i >> 7, d = i & (D - 1);
        float di = dinv[n];
        out[i] = b[d] + h[i] * (di * di);
    }
}

// ---------------- edge scatter: one wave32 per edge, float4 per lane ----------------
__global__ void __launch_bounds__(256) k_scatter(const int* __restrict__ src,
                                                 const int* __restrict__ dst,
                                                 const float* __restrict__ h,
                                                 const float* __restrict__ dinv,
                                                 float* __restrict__ out) {
    int e    = blockIdx.x * (blockDim.x >> 5) + (threadIdx.x >> 5);
    int lane = threadIdx.x & 31;
    if (e >= N_EDGES) return;

    // prefetch the source row of an edge a few waves ahead (global_prefetch_b8)
    int ep = e + 8;
    if (ep < N_EDGES)
        __builtin_prefetch(h + (size_t)src[ep] * D + lane * 4, 0, 0);

    int s = src[e], t = dst[e];
    float w = dinv[s] * dinv[t];

    const float4* hp = (const float4*)(h + (size_t)s * D);
    float4 v = hp[lane];                       // 32 lanes x 16B = full 512B row
    float* op = out + (size_t)t * D + lane * 4;
    atomicAdd(op + 0, v.x * w);
    atomicAdd(op + 1, v.y * w);
    atomicAdd(op + 2, v.z * w);
    atomicAdd(op + 3, v.w * w);
}

// ---------------- batch-norm statistics over nodes ----------------
__global__ void k_init_stats(float* __restrict__ sums, float* __restrict__ sumsq) {
    int i = threadIdx.x;
    if (i < D) { sums[i] = 0.0f; sumsq[i] = 0.0f; }
}

__global__ void __launch_bounds__(256) k_stats(const float* __restrict__ out,
                                               float* __restrict__ sums,
                                               float* __restrict__ sumsq) {
    int d = threadIdx.x & (D - 1);
    int r0 = blockIdx.x * 2 + (threadIdx.x >> 7);
    float s = 0.0f, ss = 0.0f;
    for (int r = r0; r < N_NODES; r += gridDim.x * 2) {
        float v = out[r * D + d];
        v = v > 0.0f ? v : 0.0f;               // ReLU before stats
        s += v; ss += v * v;
    }
    atomicAdd(&sums[d], s);
    atomicAdd(&sumsq[d], ss);
}

__global__ void k_bn(float* __restrict__ out, const float* __restrict__ sums,
                     const float* __restrict__ sumsq, const float* __restrict__ gamma,
                     const float* __restrict__ beta) {
    int i = blockIdx.x * blockDim.x + threadIdx.x;
    if (i >= N_NODES * D) return;
    int d = i & (D - 1);
    const float invN = 1.0f / (float)N_NODES;
    float mean = sums[d] * invN;
    float var  = sumsq[d] * invN - mean * mean;
    float v = out[i];
    v = v > 0.0f ? v : 0.0f;                   // ReLU
    out[i] = gamma[d] * (v - mean) * rsqrtf(var + BN_EPS) + beta[d];
}

// ---------------- launcher ----------------
extern "C" void kernel_launch(void* const* d_in, const int* in_sizes, int n_in,
                              void* d_out, int out_size, void* d_ws, size_t ws_size,
                              hipStream_t stream) {
    const float* x     = (const float*)d_in[0];
    const int*   edges = (const int*)  d_in[1];   // [2, E] flat
    const float* Wm    = (const float*)d_in[2];
    const float* b     = (const float*)d_in[3];
    const float* gamma = (const float*)d_in[4];
    const float* beta  = (const float*)d_in[5];
    float*       out   = (float*)d_out;

    const int* src = edges;
    const int* dst = edges + N_EDGES;

    float* deg   = (float*)d_ws;                  // N          (becomes dinv)
    float* h     = deg + N_NODES;                 // N * 128
    float* sums  = h + (size_t)N_NODES * D;       // 128
    float* sumsq = sums + D;                      // 128

    k_init_deg  <<<(N_NODES + 255) / 256, 256, 0, stream>>>(deg);
    k_deg       <<<(N_EDGES + 255) / 256, 256, 0, stream>>>(dst, deg);
    k_dinv      <<<(N_NODES + 255) / 256, 256, 0, stream>>>(deg);

    k_gemm      <<<N_NODES / 16, 256, 0, stream>>>(x, Wm, h);

    k_self_bias <<<(N_NODES * D + 255) / 256, 256, 0, stream>>>(h, deg, b, out);
    k_scatter   <<<N_EDGES / 8, 256, 0, stream>>>(src, dst, h, deg, out);

    k_init_stats<<<1, 128, 0, stream>>>(sums, sumsq);
    k_stats     <<<512, 256, 0, stream>>>(out, sums, sumsq);
    k_bn        <<<(N_NODES * D + 255) / 256, 256, 0, stream>>>(out, sums, sumsq, gamma, beta);
}